// CNN_GNN_deep_3_26448408608775
// MI455X (gfx1250) — compile-verified
//
#include <hip/hip_runtime.h>
#include <hip/hip_bf16.h>
#include <math.h>

// ---------------------------------------------------------------------------
// CNN(3d conv encoder) + 2-layer SAGE GNN, restructured for MI455X (gfx1250).
// feat[:,3:] = enc[graph_id] has only B=8 distinct rows, so segment_max over
// the 2816-dim part collapses to a 256-entry "graph-mask" table; the two big
// [16000x2819]x[2819x100] GEMMs collapse to [256x2816]x[2816x100] and
// [8x2816]x[2816x100], done with fp32 WMMA (V_WMMA_F32_16X16X4_F32).
// ---------------------------------------------------------------------------

#define B_     8
#define CIN_   25
#define NPG_   2000
#define ENC_   2816
#define HID_   100
#define KGEMM_ 2816   // GEMM K dim

typedef __attribute__((ext_vector_type(2))) float v2f;
typedef __attribute__((ext_vector_type(8))) float v8f;

// ---- monotone float <-> uint key for atomic max over signed floats --------
__device__ __forceinline__ unsigned fkey(float f) {
    unsigned u = __float_as_uint(f);
    return (u & 0x80000000u) ? ~u : (u | 0x80000000u);
}
__device__ __forceinline__ float keyf(unsigned k) {
    unsigned u = (k & 0x80000000u) ? (k ^ 0x80000000u) : ~k;
    return __uint_as_float(u);
}

// ---------------------------------------------------------------------------
// 0) zero-fill helper (agg2 / masks / counts / aggmax keys each call)
// ---------------------------------------------------------------------------
__global__ void zero_u32(unsigned* __restrict__ p, int n) {
    int i = blockIdx.x * blockDim.x + threadIdx.x;
    if (i < n) p[i] = 0u;
}

// ---------------------------------------------------------------------------
// 1) conv encoder: three direct 3d convs + relu, then 3x3x3 maxpool.
//    ~1 GMAC total: latency-bound VALU work, not worth implicit-GEMM here.
// ---------------------------------------------------------------------------
__global__ void conv1_kernel(const float* __restrict__ X, const float* __restrict__ w,
                             const float* __restrict__ b, float* __restrict__ out) {
    // X [8,25,25,6,6] -> out [8,32,21,6,6], pad D:0 H:1 W:1
    const int total = 8 * 32 * 21 * 6 * 6;
    int idx = blockIdx.x * blockDim.x + threadIdx.x;
    if (idx >= total) return;
    int x = idx % 6;  int t = idx / 6;
    int y = t % 6;    t /= 6;
    int z = t % 21;   t /= 21;
    int o = t % 32;   int bn = t / 32;
    float acc = b[o];
    for (int c = 0; c < 25; ++c)
        for (int kd = 0; kd < 5; ++kd) {
            const float* Xp = X + ((size_t)((bn * 25 + c) * 25 + (z + kd))) * 36;
            const float* Wp = w + ((size_t)((o * 25 + c) * 5 + kd)) * 9;
            for (int kh = 0; kh < 3; ++kh) {
                int yy = y + kh - 1; if (yy < 0 || yy >= 6) continue;
                for (int kw = 0; kw < 3; ++kw) {
                    int xx = x + kw - 1; if (xx < 0 || xx >= 6) continue;
                    acc = fmaf(Xp[yy * 6 + xx], Wp[kh * 3 + kw], acc);
                }
            }
        }
    out[idx] = fmaxf(acc, 0.0f);
}

__global__ void conv2_kernel(const float* __restrict__ A, const float* __restrict__ w,
                             const float* __restrict__ b, float* __restrict__ out) {
    // A [8,32,21,6,6] -> out [8,64,17,6,6], pad D:0 H:1 W:1
    const int total = 8 * 64 * 17 * 6 * 6;
    int idx = blockIdx.x * blockDim.x + threadIdx.x;
    if (idx >= total) return;
    int x = idx % 6;  int t = idx / 6;
    int y = t % 6;    t /= 6;
    int z = t % 17;   t /= 17;
    int o = t % 64;   int bn = t / 64;
    float acc = b[o];
    for (int c = 0; c < 32; ++c)
        for (int kd = 0; kd < 5; ++kd) {
            const float* Ap = A + ((size_t)((bn * 32 + c) * 21 + (z + kd))) * 36;
            const float* Wp = w + ((size_t)((o * 32 + c) * 5 + kd)) * 9;
            for (int kh = 0; kh < 3; ++kh) {
                int yy = y + kh - 1; if (yy < 0 || yy >= 6) continue;
                for (int kw = 0; kw < 3; ++kw) {
                    int xx = x + kw - 1; if (xx < 0 || xx >= 6) continue;
                    acc = fmaf(Ap[yy * 6 + xx], Wp[kh * 3 + kw], acc);
                }
            }
        }
    out[idx] = fmaxf(acc, 0.0f);
}

__global__ void conv3_kernel(const float* __restrict__ A, const float* __restrict__ w,
                             const float* __restrict__ b, float* __restrict__ out) {
    // A [8,64,17,6,6] -> out [8,64,13,4,4], VALID
    const int total = 8 * 64 * 13 * 4 * 4;
    int idx = blockIdx.x * blockDim.x + threadIdx.x;
    if (idx >= total) return;
    int x = idx % 4;  int t = idx / 4;
    int y = t % 4;    t /= 4;
    int z = t % 13;   t /= 13;
    int o = t % 64;   int bn = t / 64;
    float acc = b[o];
    for (int c = 0; c < 64; ++c)
        for (int kd = 0; kd < 5; ++kd) {
            const float* Ap = A + ((size_t)((bn * 64 + c) * 17 + (z + kd))) * 36;
            const float* Wp = w + ((size_t)((o * 64 + c) * 5 + kd)) * 9;
            for (int kh = 0; kh < 3; ++kh)
                for (int kw = 0; kw < 3; ++kw)
                    acc = fmaf(Ap[(y + kh) * 6 + (x + kw)], Wp[kh * 3 + kw], acc);
        }
    out[idx] = fmaxf(acc, 0.0f);
}

__global__ void pool_kernel(const float* __restrict__ A, float* __restrict__ enc) {
    // A [8,64,13,4,4] -> max 3x3x3 stride 1 -> [8,64,11,2,2] -> enc [8,2816]
    const int total = 8 * 64 * 11 * 2 * 2;
    int idx = blockIdx.x * blockDim.x + threadIdx.x;
    if (idx >= total) return;
    int x = idx % 2;  int t = idx / 2;
    int y = t % 2;    t /= 2;
    int d = t % 11;   t /= 11;
    int ch = t % 64;  int bn = t / 64;
    float m = -INFINITY;
    for (int dd = 0; dd < 3; ++dd)
        for (int hh = 0; hh < 3; ++hh)
            for (int ww = 0; ww < 3; ++ww) {
                float v = A[((size_t)((bn * 64 + ch) * 13 + (d + dd))) * 16 + (y + hh) * 4 + (x + ww)];
                m = fmaxf(m, v);
            }
    enc[(size_t)bn * ENC_ + ch * 44 + d * 4 + y * 2 + x] = m;
}

// ---------------------------------------------------------------------------
// 2) edge pass: per-dst neighbor-graph mask, degree count, max of node_x3
// ---------------------------------------------------------------------------
__global__ void edge_pass(const int* __restrict__ ei, const int* __restrict__ gid,
                          const float* __restrict__ x3, unsigned* __restrict__ aggk,
                          unsigned* __restrict__ mask, int* __restrict__ counts, int E) {
    int e = blockIdx.x * blockDim.x + threadIdx.x;
    if (e >= E) return;
    int s = ei[e];
    int d = ei[E + e];
    int g = gid[s];
    atomicOr(&mask[d], 1u << g);
    atomicAdd(&counts[d], 1);
    atomicMax(&aggk[(size_t)d * 3 + 0], fkey(x3[(size_t)s * 3 + 0]));
    atomicMax(&aggk[(size_t)d * 3 + 1], fkey(x3[(size_t)s * 3 + 1]));
    atomicMax(&aggk[(size_t)d * 3 + 2], fkey(x3[(size_t)s * 3 + 2]));
}

// ---------------------------------------------------------------------------
// 3) subset-DP: maxenc[mask][c] = elementwise max of enc rows in mask.
//    enc >= 0 (post-relu/maxpool), so 0 is a safe identity for row 0.
// ---------------------------------------------------------------------------
__global__ void maxenc_kernel(const float* __restrict__ enc, float* __restrict__ ME) {
    int c = blockIdx.x * blockDim.x + threadIdx.x;
    if (c >= ENC_) return;
    ME[c] = 0.0f;
    for (int m = 1; m < 256; ++m) {
        int p = m & (m - 1);
        int g = __ffs(m) - 1;
        ME[(size_t)m * ENC_ + c] = fmaxf(ME[(size_t)p * ENC_ + c], enc[(size_t)g * ENC_ + c]);
    }
}

// ---------------------------------------------------------------------------
// 4) fp32 WMMA GEMM: C[Mrows,100] = A[Mrows,2816] @ W[2816,100]
//    One wave per 16x16 output tile; K-loop of V_WMMA_F32_16X16X4_F32.
//    A frag (16x4 f32): lane = half*16+l16 holds M=l16, K = half*2 + {0,1}.
//    B frag (4x16 f32): lane holds N=l16, K = half*2 + {0,1}.
//    Out-of-range rows/cols are CLAMPED (their outputs are never stored), so
//    the inner loop is branch-free; two accumulator chains hide WMMA RAW
//    latency (D->C dependency) between back-to-back v_wmma ops.
// ---------------------------------------------------------------------------
__global__ void __launch_bounds__(32)
gemm_wmma_f32(const float* __restrict__ A, const float* __restrict__ W,
              float* __restrict__ C, int Mrows) {
    const int Ncols = HID_;
    int lane = threadIdx.x & 31;
    int half = lane >> 4;
    int l16  = lane & 15;
    int mBase = blockIdx.x * 16;
    int nBase = blockIdx.y * 16;
    int mRow = mBase + l16; if (mRow >= Mrows) mRow = Mrows - 1;  // clamp (masked at store)
    int nCol = nBase + l16;
    bool nOk = (nCol < Ncols);
    int nClamp = nOk ? nCol : (Ncols - 1);                        // clamp (masked at store)
    const float* Arow = A + (size_t)mRow * KGEMM_;
    const float* Bcol = W + nClamp;

    v8f acc0 = {}, acc1 = {};
    for (int kt = 0; kt < KGEMM_; kt += 8) {
        int k0 = kt + half * 2;       // K-block 0: k in [kt, kt+4)
        int k1 = k0 + 4;              // K-block 1: k in [kt+4, kt+8)
        v2f a0, b0, a1, b1;
        a0.x = Arow[k0];
        a0.y = Arow[k0 + 1];
        a1.x = Arow[k1];
        a1.y = Arow[k1 + 1];
        b0.x = Bcol[(size_t)k0 * Ncols];
        b0.y = Bcol[(size_t)(k0 + 1) * Ncols];
        b1.x = Bcol[(size_t)k1 * Ncols];
        b1.y = Bcol[(size_t)(k1 + 1) * Ncols];
        acc0 = __builtin_amdgcn_wmma_f32_16x16x4_f32(false, a0, false, b0,
                                                     (short)0, acc0, false, false);
        acc1 = __builtin_amdgcn_wmma_f32_16x16x4_f32(false, a1, false, b1,
                                                     (short)0, acc1, false, false);
    }
    v8f acc = acc0 + acc1;
    // C/D layout: VGPR v holds M = v (lanes 0-15) / M = v+8 (lanes 16-31), N = l16
    for (int v = 0; v < 8; ++v) {
        int m = mBase + half * 8 + v;
        if (m < Mrows && nOk) C[(size_t)m * Ncols + nCol] = acc[v];
    }
}

// ---------------------------------------------------------------------------
// 5) per-node h = SAGE1 + log_softmax, fused. One wave32 per node; lane j
//    owns columns j, j+32, j+64, j+96. Shuffle reductions for max/sum.
// ---------------------------------------------------------------------------
__global__ void node_h_kernel(const float* __restrict__ x3, const int* __restrict__ gid,
                              const unsigned* __restrict__ mask, const unsigned* __restrict__ aggk,
                              const float* __restrict__ P, const float* __restrict__ Q,
                              const float* __restrict__ Wl1, const float* __restrict__ Wr1,
                              const float* __restrict__ bl1, float* __restrict__ h, int N) {
    int wave = threadIdx.x >> 5;
    int lane = threadIdx.x & 31;
    int n = blockIdx.x * (blockDim.x >> 5) + wave;
    if (n >= N) return;

    float x0 = x3[(size_t)n * 3 + 0];
    float x1 = x3[(size_t)n * 3 + 1];
    float x2 = x3[(size_t)n * 3 + 2];
    int g = gid[n];
    unsigned mk = mask[n];
    bool has = (mk != 0u);
    float a0 = 0.f, a1 = 0.f, a2 = 0.f;
    if (has) {
        a0 = keyf(aggk[(size_t)n * 3 + 0]);
        a1 = keyf(aggk[(size_t)n * 3 + 1]);
        a2 = keyf(aggk[(size_t)n * 3 + 2]);
    }
    const float* Prow = P + (size_t)mk * HID_;
    const float* Qrow = Q + (size_t)g * HID_;

    float v[4];
    #pragma unroll
    for (int c = 0; c < 4; ++c) {
        int col = lane + 32 * c;
        if (col < HID_) {
            float t = bl1[col] + Qrow[col]
                    + x0 * Wr1[col] + x1 * Wr1[HID_ + col] + x2 * Wr1[2 * HID_ + col];
            if (has)
                t += Prow[col]
                   + a0 * Wl1[col] + a1 * Wl1[HID_ + col] + a2 * Wl1[2 * HID_ + col];
            v[c] = t;
        } else {
            v[c] = -INFINITY;
        }
    }
    // wave-wide max
    float m = fmaxf(fmaxf(v[0], v[1]), fmaxf(v[2], v[3]));
    for (int off = 16; off >= 1; off >>= 1)
        m = fmaxf(m, __shfl_xor(m, off, 32));
    // wave-wide sum of exp(v - m)
    float s = 0.f;
    #pragma unroll
    for (int c = 0; c < 4; ++c) s += __expf(v[c] - m); // exp(-inf)=0 for padded cols
    for (int off = 16; off >= 1; off >>= 1)
        s += __shfl_xor(s, off, 32);
    float lse = m + __logf(s);
    #pragma unroll
    for (int c = 0; c < 4; ++c) {
        int col = lane + 32 * c;
        if (col < HID_) h[(size_t)n * HID_ + col] = v[c] - lse;
    }
}

// ---------------------------------------------------------------------------
// 6) layer-2 mean aggregation: scatter-add h[src] row into agg2[dst].
//    h (6.4 MB) and agg2 (6.4 MB) both live in the 192 MB L2.
// ---------------------------------------------------------------------------
__global__ void edge_scatter_h(const int* __restrict__ ei, const float* __restrict__ h,
                               float* __restrict__ agg2, int E) {
    int e = blockIdx.x;
    int j = threadIdx.x;
    if (j < HID_) {
        int s = ei[e];
        int d = ei[E + e];
        atomicAdd(&agg2[(size_t)d * HID_ + j], h[(size_t)s * HID_ + j]);
    }
}

// ---------------------------------------------------------------------------
// 7) final: y = (agg2/max(cnt,1)) @ Wl2 + bl2 + h @ Wr2. Wave per node.
// ---------------------------------------------------------------------------
__global__ void node_out_kernel(const float* __restrict__ agg2, const float* __restrict__ h,
                                const int* __restrict__ counts, const float* __restrict__ Wl2,
                                const float* __restrict__ bl2, const float* __restrict__ Wr2,
                                float* __restrict__ y, int N) {
    int wave = threadIdx.x >> 5;
    int lane = threadIdx.x & 31;
    int n = blockIdx.x * (blockDim.x >> 5) + wave;
    if (n >= N) return;
    float s1 = 0.f, s2 = 0.f;
    #pragma unroll
    for (int c = 0; c < 4; ++c) {
        int col = lane + 32 * c;
        if (col < HID_) {
            s1 = fmaf(agg2[(size_t)n * HID_ + col], Wl2[col], s1);
            s2 = fmaf(h[(size_t)n * HID_ + col],   Wr2[col], s2);
        }
    }
    for (int off = 16; off >= 1; off >>= 1) {
        s1 += __shfl_xor(s1, off, 32);
        s2 += __shfl_xor(s2, off, 32);
    }
    if (lane == 0)
        y[n] = s1 / fmaxf((float)counts[n], 1.0f) + bl2[0] + s2;
}

// ---------------------------------------------------------------------------
extern "C" void kernel_launch(void* const* d_in, const int* in_sizes, int n_in,
                              void* d_out, int out_size, void* d_ws, size_t ws_size,
                              hipStream_t stream) {
    const float* X      = (const float*)d_in[0];
    const float* x3     = (const float*)d_in[1];
    const int*   ei     = (const int*)d_in[2];
    const int*   gid    = (const int*)d_in[3];
    const float* w1     = (const float*)d_in[4];
    const float* b1     = (const float*)d_in[5];
    const float* w2     = (const float*)d_in[6];
    const float* b2     = (const float*)d_in[7];
    const float* w3     = (const float*)d_in[8];
    const float* b3     = (const float*)d_in[9];
    const float* Wl1    = (const float*)d_in[10];
    const float* bl1    = (const float*)d_in[11];
    const float* Wr1    = (const float*)d_in[12];
    const float* Wl2    = (const float*)d_in[13];
    const float* bl2    = (const float*)d_in[14];
    const float* Wr2    = (const float*)d_in[15];
    float* y = (float*)d_out;

    const int E = in_sizes[2] / 2;   // 96000
    const int N = in_sizes[3];       // 16000

    // workspace carve-out (256B aligned)
    char* ws = (char*)d_ws;
    size_t off = 0;
    auto alloc = [&](size_t bytes) -> char* {
        char* p = ws + off;
        off = (off + bytes + 255) & ~(size_t)255;
        return p;
    };
    float*    act1  = (float*)alloc((size_t)8 * 32 * 21 * 36 * 4);     // conv1 out
    float*    act2  = (float*)alloc((size_t)8 * 64 * 17 * 36 * 4);     // conv2 out
    float*    act3  = (float*)alloc((size_t)8 * 64 * 13 * 16 * 4);     // conv3 out
    float*    enc   = (float*)alloc((size_t)8 * ENC_ * 4);             // [8,2816]
    float*    ME    = (float*)alloc((size_t)256 * ENC_ * 4);           // maxenc table
    float*    P     = (float*)alloc((size_t)256 * HID_ * 4);           // maxenc @ Wl1[3:]
    float*    Q     = (float*)alloc((size_t)16  * HID_ * 4);           // enc @ Wr1[3:]
    unsigned* aggk  = (unsigned*)alloc((size_t)N * 3 * 4);             // x3 max keys
    unsigned* mask  = (unsigned*)alloc((size_t)N * 4);                 // neighbor-graph mask
    int*      cnt   = (int*)alloc((size_t)N * 4);                      // degree
    float*    hbuf  = (float*)alloc((size_t)N * HID_ * 4);             // h (post log_softmax)
    float*    agg2  = (float*)alloc((size_t)N * HID_ * 4);             // layer-2 sums

    const int T = 256;
    // re-initialize accumulators every call (deterministic replay)
    zero_u32<<<(N * 3 + T - 1) / T, T, 0, stream>>>(aggk, N * 3);
    zero_u32<<<(N + T - 1) / T, T, 0, stream>>>(mask, N);
    zero_u32<<<(N + T - 1) / T, T, 0, stream>>>((unsigned*)cnt, N);
    zero_u32<<<(N * HID_ + T - 1) / T, T, 0, stream>>>((unsigned*)agg2, N * HID_);

    // conv encoder
    conv1_kernel<<<(8 * 32 * 21 * 36 + T - 1) / T, T, 0, stream>>>(X, w1, b1, act1);
    conv2_kernel<<<(8 * 64 * 17 * 36 + T - 1) / T, T, 0, stream>>>(act1, w2, b2, act2);
    conv3_kernel<<<(8 * 64 * 13 * 16 + T - 1) / T, T, 0, stream>>>(act2, w3, b3, act3);
    pool_kernel<<<(8 * 64 * 11 * 4 + T - 1) / T, T, 0, stream>>>(act3, enc);

    // graph structure pass
    edge_pass<<<(E + T - 1) / T, T, 0, stream>>>(ei, gid, x3, aggk, mask, cnt, E);

    // 256-entry mask -> maxenc table, then WMMA GEMMs
    maxenc_kernel<<<(ENC_ + 63) / 64, 64, 0, stream>>>(enc, ME);
    {
        dim3 gP(256 / 16, (HID_ + 15) / 16);       // 16 x 7 tiles
        gemm_wmma_f32<<<gP, 32, 0, stream>>>(ME, Wl1 + 3 * HID_, P, 256);
        dim3 gQ(1, (HID_ + 15) / 16);              // 1 x 7 tiles
        gemm_wmma_f32<<<gQ, 32, 0, stream>>>(enc, Wr1 + 3 * HID_, Q, 8);
    }

    // SAGE layer 1 + log_softmax (wave32 per node)
    node_h_kernel<<<(N + 7) / 8, 256, 0, stream>>>(x3, gid, mask, aggk, P, Q,
                                                   Wl1, Wr1, bl1, hbuf, N);

    // SAGE layer 2: scatter-add then per-node dots
    edge_scatter_h<<<E, 128, 0, stream>>>(ei, hbuf, agg2, E);
    node_out_kernel<<<(N + 7) / 8, 256, 0, stream>>>(agg2, hbuf, cnt, Wl2, bl2, Wr2, y, N);

    (void)n_in; (void)out_size; (void)ws_size;
}